// DataConsistencySinogram_30666066494001
// MI455X (gfx1250) — compile-verified
//
#include <hip/hip_runtime.h>
#include <hip/hip_bf16.h>

// ---------------------------------------------------------------------------
// DataConsistencySinogram on MI455X (gfx1250)
//   1) radon_kernel:  sinogram S[b*177+a][d]  (VALU + VMEM gathers)
//   2) dc_kernel:     S[.., acq] = s_target - S[.., acq]
//   3) build_h:       Toeplitz ramp-filter matrix H[k][m] = h(k-m)
//                     (mathematically identical to the reference's
//                      FFT ramp filtering: circular conv with 2*f)
//   4) filt_gemm:     SF = H x S  via V_WMMA_F32_16X16X4_F32 (f32 WMMA,
//                     one wave32 per 16x16 output tile, K-chunks of 4)
//   5) bp_kernel:     backprojection with linear detector interp
// ---------------------------------------------------------------------------

#define IMG      256
#define DETN     362
#define NACQ     45
#define NANG     177          // (45-1)*4 + 1
#define KP       364          // det padded to multiple of 4 (K of GEMM)
#define MP       368          // det padded to multiple of 16 (M of GEMM)
#define NP       368          // 2*177=354 columns padded to multiple of 16
#define PI_D     3.14159265358979323846

typedef __attribute__((ext_vector_type(2))) float v2f;
typedef __attribute__((ext_vector_type(8))) float v8f;

__device__ __forceinline__ float img_sample(const float* __restrict__ im, int yi, int xi) {
    bool valid = (xi >= 0) & (xi < IMG) & (yi >= 0) & (yi < IMG);
    int xc = xi < 0 ? 0 : (xi > IMG - 1 ? IMG - 1 : xi);
    int yc = yi < 0 ? 0 : (yi > IMG - 1 ? IMG - 1 : yi);
    float v = im[yc * IMG + xc];
    return valid ? v : 0.0f;
}

// ---- 1) Radon: one thread per (b, angle, detector bin) --------------------
__global__ void radon_kernel(const float* __restrict__ img,
                             const float* __restrict__ thetas_deg,
                             float* __restrict__ S /* [NP][KP] */) {
    int idx = blockIdx.x * blockDim.x + threadIdx.x;
    if (idx >= 2 * NANG * DETN) return;
    int d = idx % DETN;
    int t = idx / DETN;
    int a = t % NANG;
    int b = t / NANG;

    float th = thetas_deg[a] * (float)(PI_D / 180.0);
    float c = __cosf(th), s = __sinf(th);
    float T = (float)d - (float)((DETN - 1) * 0.5);   // 180.5
    const float* im = img + b * IMG * IMG;

    float acc = 0.0f;
    for (int i = 0; i < DETN; ++i) {
        float Sv = (float)i - (float)((DETN - 1) * 0.5);
        float xs = T * c - Sv * s + (float)((IMG - 1) * 0.5);
        float ys = T * s + Sv * c + (float)((IMG - 1) * 0.5);
        float x0 = floorf(xs), y0 = floorf(ys);
        float wx = xs - x0,    wy = ys - y0;
        int xi = (int)x0, yi = (int)y0;
        float v00 = img_sample(im, yi,     xi    );
        float v01 = img_sample(im, yi,     xi + 1);
        float v10 = img_sample(im, yi + 1, xi    );
        float v11 = img_sample(im, yi + 1, xi + 1);
        acc += v00 * (1.0f - wy) * (1.0f - wx)
             + v01 * (1.0f - wy) * wx
             + v10 * wy * (1.0f - wx)
             + v11 * wy * wx;
    }
    S[(b * NANG + a) * KP + d] = acc;
}

// ---- 2) Data consistency at acquired angles -------------------------------
__global__ void dc_kernel(const float* __restrict__ s_target /* [2][1][DETN][NACQ] */,
                          const int* __restrict__ acq_idx,
                          float* __restrict__ S /* [NP][KP] */) {
    int idx = blockIdx.x * blockDim.x + threadIdx.x;
    if (idx >= 2 * NACQ * DETN) return;
    int d = idx % DETN;
    int t = idx / DETN;
    int j = t % NACQ;
    int b = t / NACQ;
    int a = acq_idx[j];
    int o = (b * NANG + a) * KP + d;
    S[o] = s_target[(b * DETN + d) * NACQ + j] - S[o];
}

// ---- 3) Toeplitz ramp-filter matrix  H[k][m] = h(k - m) -------------------
// circular conv with 2*f:  h(0)=0.5, h(odd d)=-2/(pi^2 d^2), h(even)=0
__global__ void build_h(float* __restrict__ Hm /* [MP][KP] */) {
    int idx = blockIdx.x * blockDim.x + threadIdx.x;
    if (idx >= MP * KP) return;
    int m = idx % KP;
    int k = idx / KP;
    float v = 0.0f;
    if (k < DETN && m < DETN) {
        int dd = k - m;
        if (dd == 0) {
            v = 0.5f;
        } else if (dd & 1) {
            float fd = (float)dd;
            v = -2.0f / ((float)(PI_D * PI_D) * fd * fd);
        }
    }
    Hm[idx] = v;
}

// ---- 4) SF = H x S via f32 WMMA ------------------------------------------
// A fragment (16x4 f32): lanes 0-15 hold M=lane, K=k0+{0,1};
//                        lanes 16-31 hold M=lane-16, K=k0+{2,3}.
// B fragment (4x16 f32): lanes 0-15 hold N=lane, K=k0+{0,1};
//                        lanes 16-31 hold N=lane-16, K=k0+{2,3}.
// D fragment (16x16 f32): VGPR r: lanes 0-15 -> M=r, lanes 16-31 -> M=r+8;
//                        N = lane&15  => per-lane contiguous v8f along M.
__global__ void filt_gemm(const float* __restrict__ Hm /* [MP][KP] */,
                          const float* __restrict__ S  /* [NP][KP] */,
                          float* __restrict__ SF       /* [NP][MP] */) {
    int tileM = blockIdx.x;           // 0..MP/16-1
    int tileN = blockIdx.y;           // 0..NP/16-1
    int lane  = threadIdx.x;          // 0..31, wave32, EXEC all ones
    int half  = lane >> 4;            // 0 or 1
    int l     = lane & 15;
    int koff  = half * 2;

    const float* Arow = Hm + (tileM * 16 + l) * KP;
    const float* Bcol = S  + (tileN * 16 + l) * KP;

    v8f acc = {};
    #pragma unroll 4
    for (int k0 = 0; k0 < KP; k0 += 4) {
        v2f a = *(const v2f*)(Arow + k0 + koff);
        v2f b = *(const v2f*)(Bcol + k0 + koff);
        // (neg_a, A, neg_b, B, c_mod, C, reuse_a, reuse_b)
        acc = __builtin_amdgcn_wmma_f32_16x16x4_f32(
                  false, a, false, b, (short)0, acc, false, false);
    }
    float* out = SF + (tileN * 16 + l) * MP + tileM * 16 + half * 8;
    *(v8f*)out = acc;   // 32B-aligned: MP%8==0, base 256B aligned
}

// ---- helper: angle tables -------------------------------------------------
__global__ void angles_kernel(const float* __restrict__ thetas_deg,
                              float* __restrict__ ct, float* __restrict__ st) {
    int a = blockIdx.x * blockDim.x + threadIdx.x;
    if (a < NANG) {
        float th = thetas_deg[a] * (float)(PI_D / 180.0);
        ct[a] = __cosf(th);
        st[a] = __sinf(th);
    }
}

// ---- 5) Backprojection ----------------------------------------------------
__global__ void bp_kernel(const float* __restrict__ SF /* [NP][MP] */,
                          const float* __restrict__ ct,
                          const float* __restrict__ st,
                          float* __restrict__ out /* [2][1][IMG][IMG] */) {
    int idx = blockIdx.x * blockDim.x + threadIdx.x;
    if (idx >= 2 * IMG * IMG) return;
    int x = idx % IMG;
    int t = idx / IMG;
    int y = t % IMG;
    int b = t / IMG;

    float gx = (float)x - (float)((IMG - 1) * 0.5);
    float gy = (float)y - (float)((IMG - 1) * 0.5);
    const float* base = SF + b * NANG * MP;

    float acc = 0.0f;
    for (int a = 0; a < NANG; ++a) {
        float tt = ct[a] * gx + st[a] * gy + (float)((DETN - 1) * 0.5);
        float t0 = floorf(tt);
        float w  = tt - t0;
        int i0 = (int)t0;
        int i1 = i0 + 1;
        const float* col = base + a * MP;
        float v0 = (i0 >= 0 && i0 < DETN) ? col[i0] : 0.0f;
        float v1 = (i1 >= 0 && i1 < DETN) ? col[i1] : 0.0f;
        acc += v0 * (1.0f - w) + v1 * w;
    }
    out[idx] = acc * (float)(PI_D / (2.0 * (double)NANG));
}

// ---------------------------------------------------------------------------
extern "C" void kernel_launch(void* const* d_in, const int* in_sizes, int n_in,
                              void* d_out, int out_size, void* d_ws, size_t ws_size,
                              hipStream_t stream) {
    const float* x_source   = (const float*)d_in[0];   // (2,1,256,256)
    const float* s_target   = (const float*)d_in[1];   // (2,1,362,45)
    const float* thetas_deg = (const float*)d_in[2];   // (177,)
    const int*   acq_idx    = (const int*)d_in[3];     // (45,)
    float*       out        = (float*)d_out;           // (2,1,256,256)

    // workspace layout (256B-aligned slabs)
    char* ws = (char*)d_ws;
    const size_t SINO_BYTES = (size_t)NP * KP * sizeof(float);   // 535808
    const size_t H_BYTES    = (size_t)MP * KP * sizeof(float);   // 535808
    const size_t SF_BYTES   = (size_t)NP * MP * sizeof(float);   // 541696
    float* SINO = (float*)(ws);
    float* Hm   = (float*)(ws + SINO_BYTES);
    float* SF   = (float*)(ws + SINO_BYTES + H_BYTES);
    float* ct   = (float*)(ws + SINO_BYTES + H_BYTES + SF_BYTES);
    float* st   = (float*)(ws + SINO_BYTES + H_BYTES + SF_BYTES + 1024);

    // zero padded sinogram (graph-capturable memset)
    hipMemsetAsync(SINO, 0, SINO_BYTES, stream);

    {   // ramp-filter Toeplitz matrix
        int n = MP * KP;
        build_h<<<(n + 255) / 256, 256, 0, stream>>>(Hm);
    }
    angles_kernel<<<1, 256, 0, stream>>>(thetas_deg, ct, st);
    {   // radon
        int n = 2 * NANG * DETN;
        radon_kernel<<<(n + 255) / 256, 256, 0, stream>>>(x_source, thetas_deg, SINO);
    }
    {   // data consistency
        int n = 2 * NACQ * DETN;
        dc_kernel<<<(n + 255) / 256, 256, 0, stream>>>(s_target, acq_idx, SINO);
    }
    {   // WMMA GEMM: SF = H x S
        dim3 grid(MP / 16, NP / 16);
        filt_gemm<<<grid, 32, 0, stream>>>(Hm, SINO, SF);
    }
    {   // backprojection
        int n = 2 * IMG * IMG;
        bp_kernel<<<(n + 255) / 256, 256, 0, stream>>>(SF, ct, st, out);
    }
}